// GANTGNN_13005160972783
// MI455X (gfx1250) — compile-verified
//
#include <hip/hip_runtime.h>

typedef __attribute__((ext_vector_type(2))) float v2f;
typedef __attribute__((ext_vector_type(4))) float f32x4;
typedef __attribute__((ext_vector_type(8))) float v8f;

#define N_NODES 6144
#define F_DIM   256
#define KC      128            // adj K-chunk per LDS stage
#define NCHUNK  (N_NODES / KC) // 48
#define ROWS    32             // rows per workgroup (2 M-tiles)
#define ASTR    132            // padded LDS stride for adj chunk (bank-conflict free)
#define OSTR    264            // padded LDS stride for output staging

// ---------------------------------------------------------------------------
// Kernel 1: h = x @ W   (6144x256 @ 256x256), fp32 WMMA 16x16x4.
// One wave per 16x16 output tile; 8 waves (256 threads) per block.
// A-frag (32b A 16x4): lanes0-15 row=l16 hold K=k,k+1; lanes16-31 hold K=k+2,k+3.
// B-frag mirrors A with lane = column.
// ---------------------------------------------------------------------------
__global__ __launch_bounds__(256) void gemm_xw(const float* __restrict__ x,
                                               const float* __restrict__ W,
                                               float* __restrict__ h) {
    const int lane = threadIdx.x & 31;
    const int widx = threadIdx.x >> 5;
    const int gw   = blockIdx.x * 8 + widx;       // 0..6143 tiles
    const int tm   = gw >> 4;                     // row tile 0..383
    const int tn   = gw & 15;                     // col tile 0..15
    const int row0 = tm * 16;
    const int col0 = tn * 16;
    const int hi   = (lane >= 16) ? 1 : 0;
    const int l16  = lane & 15;

    v8f c = {};
    const float* xrow = x + (size_t)(row0 + l16) * F_DIM + 2 * hi;
    const int    bcol = col0 + l16;

    for (int k = 0; k < F_DIM; k += 4) {
        v2f a;
        a.x = xrow[k];
        a.y = xrow[k + 1];
        const int kb = k + 2 * hi;
        v2f b;
        b.x = W[(size_t)kb * F_DIM + bcol];
        b.y = W[(size_t)(kb + 1) * F_DIM + bcol];
        c = __builtin_amdgcn_wmma_f32_16x16x4_f32(false, a, false, b,
                                                  (short)0, c, false, false);
    }
    float* o = h + (size_t)(row0 + 8 * hi) * F_DIM + col0 + l16;
#pragma unroll
    for (int v = 0; v < 8; ++v) o[(size_t)v * F_DIM] = c[v];
}

// ---------------------------------------------------------------------------
// Kernel 2: agg = adj @ h with fused row-degree, then out = LN(agg/deg)*g + b.
// One block (8 waves, 256 thr) per 32-row slab. Wave w computes a 32x32 patch:
// 2 M-tiles x 2 N-tiles (cols 32w..32w+31), so each B fragment feeds 2 WMMAs.
// adj is double-buffered through LDS with non-temporal streaming loads;
// h (6 MB) stays L2-resident and is read per-fragment.
// ---------------------------------------------------------------------------
__global__ __launch_bounds__(256) void gat_agg_ln(const float* __restrict__ adj,
                                                  const float* __restrict__ h,
                                                  const float* __restrict__ gamma,
                                                  const float* __restrict__ beta,
                                                  float* __restrict__ out) {
    __shared__ float sA[2][ROWS * ASTR];   // 2 x (32 x 128) adj chunk, padded
    __shared__ float sOut[ROWS * OSTR];    // 32 x 256 output staging, padded
    __shared__ float sRedA[ROWS * 8];
    __shared__ float sRedB[ROWS * 8];
    __shared__ float sInv[ROWS];

    const int tid  = threadIdx.x;
    const int lane = tid & 31;
    const int widx = tid >> 5;             // wave 0..7
    const int l16  = lane & 15;
    const int hi   = (lane >= 16) ? 1 : 0;
    const int row0 = blockIdx.x * ROWS;

    // cooperative loader: thread -> (row 0..31, 16-float slice 0..7)
    const int lr  = tid >> 3;              // 0..31
    const int lc8 = tid & 7;               // 0..7
    const float* aptr = adj + (size_t)(row0 + lr) * N_NODES + lc8 * 16;

    const int colA = widx * 32 + l16;
    const int colB = colA + 16;

    float degsum = 0.f;
    v8f c00 = {}, c01 = {}, c10 = {}, c11 = {};

    // ---- prologue: stage chunk 0 ----
    {
        const f32x4* p = (const f32x4*)aptr;
        f32x4 q0 = __builtin_nontemporal_load(p + 0);
        f32x4 q1 = __builtin_nontemporal_load(p + 1);
        f32x4 q2 = __builtin_nontemporal_load(p + 2);
        f32x4 q3 = __builtin_nontemporal_load(p + 3);
        degsum += q0.x + q0.y + q0.z + q0.w + q1.x + q1.y + q1.z + q1.w +
                  q2.x + q2.y + q2.z + q2.w + q3.x + q3.y + q3.z + q3.w;
        f32x4* d = (f32x4*)&sA[0][lr * ASTR + lc8 * 16];
        d[0] = q0; d[1] = q1; d[2] = q2; d[3] = q3;
    }
    __syncthreads();

    for (int c = 0; c < NCHUNK; ++c) {
        const int kb = c * KC;

        // ---- issue next chunk's streaming loads (overlap with compute) ----
        f32x4 q0, q1, q2, q3;
        const bool more = (c + 1 < NCHUNK);
        if (more) {
            const f32x4* p = (const f32x4*)(aptr + kb + KC);
            q0 = __builtin_nontemporal_load(p + 0);
            q1 = __builtin_nontemporal_load(p + 1);
            q2 = __builtin_nontemporal_load(p + 2);
            q3 = __builtin_nontemporal_load(p + 3);
            degsum += q0.x + q0.y + q0.z + q0.w + q1.x + q1.y + q1.z + q1.w +
                      q2.x + q2.y + q2.z + q2.w + q3.x + q3.y + q3.z + q3.w;
        }
        if (c + 2 < NCHUNK) __builtin_prefetch(aptr + kb + 2 * KC, 0, 0);

        // ---- 32 WMMA K-steps over the current chunk ----
        const float* sa     = sA[c & 1];
        const float* aBase0 = sa + l16 * ASTR + 2 * hi;        // M-tile 0
        const float* aBase1 = aBase0 + 16 * ASTR;              // M-tile 1
        for (int kk = 0; kk < KC; kk += 4) {
            v2f a0; a0.x = aBase0[kk]; a0.y = aBase0[kk + 1];
            v2f a1; a1.x = aBase1[kk]; a1.y = aBase1[kk + 1];
            const int kg = kb + kk + 2 * hi;
            const float* hr0 = h + (size_t)kg * F_DIM;
            const float* hr1 = hr0 + F_DIM;
            v2f b0; b0.x = hr0[colA]; b0.y = hr1[colA];
            v2f b1; b1.x = hr0[colB]; b1.y = hr1[colB];
            c00 = __builtin_amdgcn_wmma_f32_16x16x4_f32(false, a0, false, b0,
                                                        (short)0, c00, false, false);
            c01 = __builtin_amdgcn_wmma_f32_16x16x4_f32(false, a0, false, b1,
                                                        (short)0, c01, false, false);
            c10 = __builtin_amdgcn_wmma_f32_16x16x4_f32(false, a1, false, b0,
                                                        (short)0, c10, false, false);
            c11 = __builtin_amdgcn_wmma_f32_16x16x4_f32(false, a1, false, b1,
                                                        (short)0, c11, false, false);
        }
        __syncthreads();

        // ---- commit next chunk into the other LDS buffer ----
        if (more) {
            f32x4* d = (f32x4*)&sA[(c + 1) & 1][lr * ASTR + lc8 * 16];
            d[0] = q0; d[1] = q1; d[2] = q2; d[3] = q3;
        }
        __syncthreads();
    }

    // ---- degree reduction: 8 partials per row ----
    sRedA[lr * 8 + lc8] = degsum;
    __syncthreads();
    if (tid < ROWS) {
        float s = 0.f;
#pragma unroll
        for (int i = 0; i < 8; ++i) s += sRedA[tid * 8 + i];
        sInv[tid] = 1.0f / s;              // deg >= 1 (self loop guaranteed)
    }
    // ---- stage accumulators to LDS (concurrent with degree reduce) ----
    {
        float* o00 = &sOut[(8 * hi) * OSTR + colA];
        float* o01 = &sOut[(8 * hi) * OSTR + colB];
        float* o10 = &sOut[(16 + 8 * hi) * OSTR + colA];
        float* o11 = &sOut[(16 + 8 * hi) * OSTR + colB];
#pragma unroll
        for (int v = 0; v < 8; ++v) {
            o00[v * OSTR] = c00[v];
            o01[v * OSTR] = c01[v];
            o10[v * OSTR] = c10[v];
            o11[v * OSTR] = c11[v];
        }
    }
    __syncthreads();

    // ---- LayerNorm: thread handles 32 features of row lr ----
    const float invdeg = sInv[lr];
    float vals[32];
    float s1 = 0.f, s2 = 0.f;
#pragma unroll
    for (int i = 0; i < 32; ++i) {
        float v = sOut[lr * OSTR + lc8 * 32 + i] * invdeg;
        vals[i] = v;
        s1 += v;
        s2 += v * v;
    }
    sRedA[lr * 8 + lc8] = s1;
    sRedB[lr * 8 + lc8] = s2;
    __syncthreads();
    float su = 0.f, sq = 0.f;
#pragma unroll
    for (int i = 0; i < 8; ++i) {
        su += sRedA[lr * 8 + i];
        sq += sRedB[lr * 8 + i];
    }
    const float mu  = su * (1.0f / 256.0f);
    const float var = sq * (1.0f / 256.0f) - mu * mu;
    const float rs  = rsqrtf(var + 1e-5f);
    float* op = out + (size_t)(row0 + lr) * F_DIM + lc8 * 32;
#pragma unroll
    for (int i = 0; i < 32; ++i) {
        const int f = lc8 * 32 + i;
        op[i] = (vals[i] - mu) * rs * gamma[f] + beta[f];
    }
}

extern "C" void kernel_launch(void* const* d_in, const int* in_sizes, int n_in,
                              void* d_out, int out_size, void* d_ws, size_t ws_size,
                              hipStream_t stream) {
    const float* x     = (const float*)d_in[0];   // (6144, 256)
    const float* adj   = (const float*)d_in[1];   // (6144, 6144)
    const float* W     = (const float*)d_in[2];   // (256, 256)
    // d_in[3] = a : provably unused (softmax rows are constant over unmasked entries)
    const float* gamma = (const float*)d_in[4];   // (256,)
    const float* beta  = (const float*)d_in[5];   // (256,)
    float* out = (float*)d_out;                   // (6144, 256)
    float* h   = (float*)d_ws;                    // 6144*256 f32 = 6 MB scratch

    gemm_xw<<<768, 256, 0, stream>>>(x, W, h);
    gat_agg_ln<<<N_NODES / ROWS, 256, 0, stream>>>(adj, h, gamma, beta, out);
}